// Shiftwindow_attention_66322884985157
// MI455X (gfx1250) — compile-verified
//
#include <hip/hip_runtime.h>
#include <math.h>

typedef __attribute__((ext_vector_type(2))) float v2f;
typedef __attribute__((ext_vector_type(8))) float v8f;
typedef __attribute__((ext_vector_type(4))) float vf4;
typedef __attribute__((ext_vector_type(4))) int   v4i;

typedef __attribute__((address_space(1))) v4i g_v4i;   // global int4
typedef __attribute__((address_space(3))) v4i l_v4i;   // LDS int4

#define CHANNELS 512
#define HW 16384      // 128*128
#define HWSHIFT 14
#define POSB 128      // positions per block (phase 1)
#define CHUNK 8       // channels per LDS chunk (phase 1)

#if defined(__has_builtin)
#  if __has_builtin(__builtin_amdgcn_global_load_async_to_lds_b128)
#    define USE_ASYNC_LDS 1
#  else
#    define USE_ASYNC_LDS 0
#  endif
#  if __has_builtin(__builtin_amdgcn_s_wait_asynccnt)
#    define HAVE_WAIT_ASYNC 1
#  else
#    define HAVE_WAIT_ASYNC 0
#  endif
#else
#  define USE_ASYNC_LDS 0
#  define HAVE_WAIT_ASYNC 0
#endif

// Copy 16 bytes global -> LDS. Async (ASYNCcnt) when the gfx1250 builtin
// exists; otherwise synchronous b128 load + ds_store with the same structure.
__device__ __forceinline__ void copy16_g2l(const float* g, float* l) {
#if USE_ASYNC_LDS
  __builtin_amdgcn_global_load_async_to_lds_b128(
      (g_v4i*)(void*)g, (l_v4i*)(void*)l, 0, 0);
#else
  *(vf4*)l = *(const vf4*)g;
#endif
}

__device__ __forceinline__ void wait_async(int n) {
#if USE_ASYNC_LDS
#  if HAVE_WAIT_ASYNC
  if (n == 0) __builtin_amdgcn_s_wait_asynccnt(0);
  else        __builtin_amdgcn_s_wait_asynccnt(1);
#  else
  if (n == 0) asm volatile("s_wait_asynccnt 0x0" ::: "memory");
  else        asm volatile("s_wait_asynccnt 0x1" ::: "memory");
#  endif
#endif
  (void)n;
}

// ---------------------------------------------------------------------------
// Phase 1: x_dot[b,hw] = sum_c in[b,c,hw]*w[c]   via V_WMMA_F32_16X16X4_F32.
//
// Memory path: per 8-channel chunk, wave w async-copies channel (c0+w)'s
// 128-position row (512 B, one GLOBAL_LOAD_ASYNC_TO_LDS_B128) into a
// double-buffered LDS tile [2][8][128].  Compute path: each wave owns a
// 16-position sub-tile and feeds WMMA operands from LDS:
//   A(16x4) = w[c0..c0+3] replicated down rows (lanes=M; half-waves split K)
//   B(4x16) = lds[k][wave*16+n]                (lanes=N; same K split)
// D row 0 -> acc[0] lanes 0-15 = the 16 dot products.
// ---------------------------------------------------------------------------
__global__ void __launch_bounds__(256)
p1_conv_wmma(const float* __restrict__ in, const float* __restrict__ w,
             float* __restrict__ xdot) {
  __shared__ float lw[CHANNELS];
  __shared__ float buf[2][CHUNK][POSB];

  const int tid  = threadIdx.x;
  const int lane = tid & 31;
  const int wave = tid >> 5;

  lw[tid]       = w[tid];
  lw[tid + 256] = w[tid + 256];

  const int b   = blockIdx.x >> 7;           // 128 position-tiles per batch
  const int hw0 = (blockIdx.x & 127) << 7;   // 128 positions per block
  const float* src = in + (long long)b * CHANNELS * HW + hw0;

  const int n  = lane & 15;                  // B column within wave's sub-tile
  const int kb = (lane >> 4) << 1;           // K base for this half-wave: 0/2
  const int col = (wave << 4) + n;           // position column in block tile

  // prologue: chunk 0 -> buf[0]
  copy16_g2l(src + (long long)wave * HW + (lane << 2), &buf[0][wave][lane << 2]);
  __syncthreads();                           // also publishes lw[]

  v8f acc0 = {};
  v8f acc1 = {};
  for (int ci = 0; ci < CHANNELS / CHUNK; ++ci) {
    const int c0 = ci * CHUNK;
    const int cur = ci & 1;
    if (ci + 1 < CHANNELS / CHUNK) {         // issue next chunk into other buf
      copy16_g2l(src + (long long)(c0 + CHUNK + wave) * HW + (lane << 2),
                 &buf[cur ^ 1][wave][lane << 2]);
      wait_async(1);                         // own chunk-ci copy complete
    } else {
      wait_async(0);
    }
    __syncthreads();                         // all waves' chunk-ci data ready

    v2f a0, a1, b0, b1;
    a0.x = lw[c0 + kb];     a0.y = lw[c0 + kb + 1];
    a1.x = lw[c0 + 4 + kb]; a1.y = lw[c0 + 4 + kb + 1];
    b0.x = buf[cur][kb][col];     b0.y = buf[cur][kb + 1][col];
    b1.x = buf[cur][4 + kb][col]; b1.y = buf[cur][5 + kb][col];
    acc0 = __builtin_amdgcn_wmma_f32_16x16x4_f32(false, a0, false, b0,
                                                 (short)0, acc0, false, false);
    acc1 = __builtin_amdgcn_wmma_f32_16x16x4_f32(false, a1, false, b1,
                                                 (short)0, acc1, false, false);
    __syncthreads();                         // buf[cur] free for reuse
  }

  const float r = acc0[0] + acc1[0];
  if (lane < 16)
    xdot[(long long)b * HW + hw0 + (wave << 4) + lane] = r;
}

// ---------------------------------------------------------------------------
// Phase 2: per (b, mask k) windowed softmax stats, then att1 = 1 + sum_k p_k.
// For each shift i, the 4 masks partition HxW; membership = 2 interval tests.
// x_dot is 512 KB -> L2 resident; this kernel is negligible time.
// ---------------------------------------------------------------------------
__global__ void __launch_bounds__(256)
p2_softmax(const float* __restrict__ xdot, float* __restrict__ att1) {
  const int b = blockIdx.x;
  const float* x = xdot + (long long)b * HW;
  __shared__ float part[16][256];
  __shared__ float sM[16];
  __shared__ float sS[16];
  const int tid = threadIdx.x;

  // ---- pass 1: per-window max ----
  float mx[16];
#pragma unroll
  for (int k = 0; k < 16; ++k) mx[k] = -__builtin_inff();
  for (int p = tid; p < HW; p += 256) {
    const float v = x[p];
    const int h = p >> 7, wv = p & 127;
#pragma unroll
    for (int i = 0; i < 4; ++i) {
      const int sft = 16 * i;
      const bool inh = (h >= sft) && (h < 64 + sft);
      const bool inw = (wv >= sft) && (wv < 64 + sft);
      const int m = (inh ? 0 : 2) + (inw ? 0 : 1);
#pragma unroll
      for (int mm = 0; mm < 4; ++mm)
        mx[4 * i + mm] = (m == mm) ? fmaxf(mx[4 * i + mm], v) : mx[4 * i + mm];
    }
  }
#pragma unroll
  for (int k = 0; k < 16; ++k) part[k][tid] = mx[k];
  __syncthreads();
  if (tid < 16) {
    float m = -__builtin_inff();
    for (int j = 0; j < 256; ++j) m = fmaxf(m, part[tid][j]);
    sM[tid] = m;
  }
  __syncthreads();

  // ---- pass 2: per-window sum of exp ----
  float sm[16];
#pragma unroll
  for (int k = 0; k < 16; ++k) sm[k] = 0.f;
  for (int p = tid; p < HW; p += 256) {
    const float v = x[p];
    const int h = p >> 7, wv = p & 127;
#pragma unroll
    for (int i = 0; i < 4; ++i) {
      const int sft = 16 * i;
      const bool inh = (h >= sft) && (h < 64 + sft);
      const bool inw = (wv >= sft) && (wv < 64 + sft);
      const int m = (inh ? 0 : 2) + (inw ? 0 : 1);
      const float e = __expf(v - sM[4 * i + m]);
#pragma unroll
      for (int mm = 0; mm < 4; ++mm)
        sm[4 * i + mm] += (m == mm) ? e : 0.f;
    }
  }
#pragma unroll
  for (int k = 0; k < 16; ++k) part[k][tid] = sm[k];
  __syncthreads();
  if (tid < 16) {
    float s = 0.f;
    for (int j = 0; j < 256; ++j) s += part[tid][j];
    sS[tid] = 1.0f / s;
  }
  __syncthreads();

  // ---- pass 3: att1 = 1 + sum over the 4 owning windows ----
  for (int p = tid; p < HW; p += 256) {
    const float v = x[p];
    const int h = p >> 7, wv = p & 127;
    float att = 0.f;
#pragma unroll
    for (int i = 0; i < 4; ++i) {
      const int sft = 16 * i;
      const bool inh = (h >= sft) && (h < 64 + sft);
      const bool inw = (wv >= sft) && (wv < 64 + sft);
      const int k = 4 * i + (inh ? 0 : 2) + (inw ? 0 : 1);
      att += __expf(v - sM[k]) * sS[k];
    }
    att1[(long long)b * HW + p] = 1.0f + att;
  }
}

// ---------------------------------------------------------------------------
// Phase 3: out[b,c,hw] = in[b,c,hw] * att1[b,hw].  Pure 512 MB stream:
// 128-bit vector loads/stores, non-temporal on the big streams so the
// 192 MB L2 keeps att1 (512 KB) hot instead.
// ---------------------------------------------------------------------------
__global__ void __launch_bounds__(256)
p3_scale(const float* __restrict__ in, const float* __restrict__ att1,
         float* __restrict__ out) {
  const long long i4 = (long long)blockIdx.x * 256 + threadIdx.x; // one float4
  const long long e  = i4 << 2;                 // flat element index
  const int bb = (int)(e >> 23);                // / (512*16384)
  const int hw = (int)(e & (HW - 1));
  const vf4* iv = (const vf4*)in;
  const vf4* av = (const vf4*)att1;
  vf4* ov = (vf4*)out;
  const vf4 s = av[((long long)bb << 12) + (hw >> 2)];
  const vf4 v = __builtin_nontemporal_load(&iv[i4]);
  __builtin_nontemporal_store(v * s, &ov[i4]);
}

extern "C" void kernel_launch(void* const* d_in, const int* in_sizes, int n_in,
                              void* d_out, int out_size, void* d_ws, size_t ws_size,
                              hipStream_t stream) {
  const float* in = (const float*)d_in[0];   // [8,512,128,128] f32
  const float* w  = (const float*)d_in[1];   // [512] f32
  float* out  = (float*)d_out;               // [8,512,128,128] f32
  float* xdot = (float*)d_ws;                // 8*16384 f32  (512 KB)
  float* att1 = xdot + 8 * HW;               // 8*16384 f32  (512 KB)

  // 1024 blocks x 256 threads: each block = 128 positions x 512 channels.
  p1_conv_wmma<<<1024, 256, 0, stream>>>(in, w, xdot);
  p2_softmax<<<8, 256, 0, stream>>>(xdot, att1);
  // 67,108,864 elements / 4 per thread-float4 = 16,777,216 threads.
  p3_scale<<<65536, 256, 0, stream>>>(in, att1, out);
}